// BipartiteNANDGraphLayer_15573551415963
// MI455X (gfx1250) — compile-verified
//
#include <hip/hip_runtime.h>
#include <cstdint>

#define BB 64
#define OO 4096
#define II 4096
#define WW 128   // int32 words per row (512 bytes)

typedef __attribute__((ext_vector_type(4))) unsigned int v4u;
typedef __attribute__((ext_vector_type(4))) int          v4i;
typedef __attribute__((ext_vector_type(8))) int          v8i;

__device__ __forceinline__ int rfl(int v) { return __builtin_amdgcn_readfirstlane(v); }

// Counter-based hash -> uniform [0,1)
__device__ __forceinline__ float u01(uint32_t a, uint32_t b, uint32_t c) {
    uint32_t x = a * 0x9E3779B9u + b * 0x85EBCA6Bu + c * 0xC2B2AE35u + 0x27220A95u;
    x ^= x >> 16; x *= 0x7FEB352Du;
    x ^= x >> 15; x *= 0x846CA68Bu;
    x ^= x >> 16;
    return (float)(x >> 8) * (1.0f / 16777216.0f);
}

// ---------------------------------------------------------------------------
// Phase 1a: per logit row (2*O rows of I floats): row max, normalized 32-chunk
// inclusive CDF (chunk = 128 contiguous elements), 1/sum. One wave per row.
// ---------------------------------------------------------------------------
__global__ __launch_bounds__(256) void row_stats_kernel(
    const float* __restrict__ adj, float* __restrict__ cdf,
    float* __restrict__ rowmax, float* __restrict__ invtot)
{
    const int lane = threadIdx.x & 31;
    const int rid  = blockIdx.x * 8 + (threadIdx.x >> 5);
    const float* row = adj + (size_t)rid * II;

    float m = -3.4e38f;
    for (int i = lane; i < II; i += 32) m = fmaxf(m, row[i]);
    for (int d = 16; d; d >>= 1) m = fmaxf(m, __shfl_xor(m, d, 32));

    float s = 0.0f;
    const int c0 = lane * 128;
    for (int i = c0; i < c0 + 128; ++i) s += __expf(row[i] - m);

    float ps = s;                                  // inclusive scan over lanes
    for (int d = 1; d < 32; d <<= 1) {
        float n = __shfl_up(ps, d, 32);
        if (lane >= d) ps += n;
    }
    float tot = __shfl(ps, 31, 32);

    cdf[(size_t)rid * 32 + lane] = ps / tot;
    if (lane == 0) { rowmax[rid] = m; invtot[rid] = 1.0f / tot; }
}

// ---------------------------------------------------------------------------
// Phase 1b: one thread per (b,o): inverse-CDF categorical draw x2 + Bernoulli.
// ---------------------------------------------------------------------------
__global__ __launch_bounds__(256) void sample_kernel(
    const float* __restrict__ adj, const float* __restrict__ invlg,
    const float* __restrict__ cdf, const float* __restrict__ rowmax,
    const float* __restrict__ invtot,
    int* __restrict__ conn_out, int* __restrict__ inv_out)
{
    const int gid = blockIdx.x * blockDim.x + threadIdx.x;
    if (gid >= BB * OO) return;
    const int b = gid / OO, o = gid % OO;

    for (int f = 0; f < 2; ++f) {
        const int rid = f * OO + o;
        const float u = u01((uint32_t)b, (uint32_t)o, (uint32_t)f);

        const float* pc = cdf + (size_t)rid * 32;
        int   c    = 31;
        float base = 0.0f, prev = 0.0f;
        for (int j = 0; j < 32; ++j) {
            float v = pc[j];
            if (u < v) { c = j; base = prev; break; }
            prev = v;
        }

        const float* row = adj + (size_t)rid * II;
        const float  m  = rowmax[rid];
        const float  it = invtot[rid];
        int   idx = c * 128 + 127;
        float acc = base;
        for (int i = c * 128; i < c * 128 + 128; ++i) {
            acc += __expf(row[i] - m) * it;
            if (u < acc) { idx = i; break; }
        }
        conn_out[(size_t)gid * 2 + f] = idx;
    }

    const float u2 = u01((uint32_t)b, (uint32_t)o, 2u);
    const float p  = 1.0f / (1.0f + __expf(-invlg[o]));
    inv_out[gid] = (u2 < p) ? 1 : 0;
}

// ---------------------------------------------------------------------------
// Phase 2: TDM-gathered NAND/NOR. One workgroup = one batch x 64 outputs.
// Wave 0 issues 8 tensor_load_to_lds gather ops (16 rows x 512B each -> 64KB
// LDS), waits TENSORcnt, barrier; 8 waves compute ~(a|b)/~(a&b) from LDS and
// store coalesced b128 to global.
// ---------------------------------------------------------------------------
__global__ __launch_bounds__(256) void gather_nand_kernel(
    const int* __restrict__ bits, const int* __restrict__ conn,
    const int* __restrict__ invm, int* __restrict__ out)
{
    __shared__ int lds_rows[64 * 2 * WW];          // 64 KB: 128 gathered rows
    const int b   = blockIdx.y;
    const int o0  = blockIdx.x * 64;
    const int tid = threadIdx.x;

    if (tid < 32) {
        const uint64_t gaddr = (uint64_t)(uintptr_t)(bits + (size_t)b * II * WW);
        const uint32_t lds0  = (uint32_t)(uintptr_t)(&lds_rows[0]);
        for (int t = 0; t < 8; ++t) {
            // 16 gathered row indices for outputs o0+8t .. o0+8t+7 (2 each),
            // packed two 16-bit indices per dword, forced uniform (SGPR).
            const int cbase = ((b * OO) + (o0 + t * 8)) * 2;
            int d0 = rfl(conn[cbase +  0] | (conn[cbase +  1] << 16));
            int d1 = rfl(conn[cbase +  2] | (conn[cbase +  3] << 16));
            int d2 = rfl(conn[cbase +  4] | (conn[cbase +  5] << 16));
            int d3 = rfl(conn[cbase +  6] | (conn[cbase +  7] << 16));
            int d4 = rfl(conn[cbase +  8] | (conn[cbase +  9] << 16));
            int d5 = rfl(conn[cbase + 10] | (conn[cbase + 11] << 16));
            int d6 = rfl(conn[cbase + 12] | (conn[cbase + 13] << 16));
            int d7 = rfl(conn[cbase + 14] | (conn[cbase + 15] << 16));

            // D# group0: count=1, gather_mode=1 (16-bit idx), lds_addr,
            // 57-bit global_addr, type=2 ("image").
            v4u g0 = { 1u | (1u << 31),
                       lds0 + (uint32_t)t * 16u * WW * 4u,
                       (uint32_t)(gaddr & 0xffffffffu),
                       (uint32_t)(gaddr >> 32) | (2u << 30) };
            // D# group1: data_size=4B(code 2); tensor_dim0=128 dwords;
            // tensor_dim1=4096 rows; tile_dim0=128; tile_dim1=16 valid
            // indices; tensor_dim0_stride=128.
            v8i g1 = { 0x20000, (128 << 16), (4096 << 16), (128 << 16),
                       16, 128, 0, 0 };
            v4i g2 = { d0, d1, d2, d3 };           // row_index_0..7
            v4i g3 = { d4, d5, d6, d7 };           // row_index_8..15
            v8i g4 = { 0, 0, 0, 0, 0, 0, 0, 0 };   // extra group (this toolchain's 6-arg form)
            __builtin_amdgcn_tensor_load_to_lds(g0, g1, g2, g3, g4, 0);
        }
        __builtin_amdgcn_s_wait_tensorcnt(0);      // all 8 TDM gathers done
    }
    __syncthreads();

    const int lane = tid & 31;
    const int wv   = tid >> 5;
    for (int r = 0; r < 8; ++r) {
        const int k = wv * 8 + r;                  // output within tile
        const v4i a = *(const v4i*)&lds_rows[(2 * k    ) * WW + lane * 4];
        const v4i c = *(const v4i*)&lds_rows[(2 * k + 1) * WW + lane * 4];
        const int iv = invm[b * OO + o0 + k];
        const v4i res = iv ? ~(a | c) : ~(a & c);  // NOR : NAND
        *(v4i*)&out[(((size_t)b * OO) + o0 + k) * WW + lane * 4] = res;
    }
}

// ---------------------------------------------------------------------------
extern "C" void kernel_launch(void* const* d_in, const int* in_sizes, int n_in,
                              void* d_out, int out_size, void* d_ws, size_t ws_size,
                              hipStream_t stream)
{
    (void)in_sizes; (void)n_in; (void)out_size; (void)ws_size;

    const int*   bits  = (const int*)d_in[0];     // (B, I, W) int32
    const float* adj   = (const float*)d_in[1];   // (2, O, I) f32
    const float* invlg = (const float*)d_in[2];   // (O,) f32

    // d_out layout (return order): function_outputs (B*O*W), conn (B*O*2), invert (B*O)
    int* out      = (int*)d_out;
    int* conn_out = out + (size_t)BB * OO * WW;
    int* inv_out  = conn_out + (size_t)BB * OO * 2;

    // workspace: normalized chunk CDFs + per-row max + 1/sum (~1.1 MB)
    float* cdf    = (float*)d_ws;                 // 2*O*32
    float* rowmax = cdf + (size_t)2 * OO * 32;    // 2*O
    float* invtot = rowmax + (size_t)2 * OO;      // 2*O

    row_stats_kernel<<<2 * OO / 8, 256, 0, stream>>>(adj, cdf, rowmax, invtot);
    sample_kernel<<<(BB * OO + 255) / 256, 256, 0, stream>>>(
        adj, invlg, cdf, rowmax, invtot, conn_out, inv_out);
    dim3 grid(OO / 64, BB);
    gather_nand_kernel<<<grid, 256, 0, stream>>>(bits, conn_out, inv_out, out);
}